// ProCoSLoss_52295521796889
// MI455X (gfx1250) — compile-verified
//
#include <hip/hip_runtime.h>
#include <math.h>

// Problem constants (from the reference)
#define P_DIM     128
#define NU        63          // P/2 - 1
#define C_CLS     1000
#define N_BATCH   512
#define KAPPA_MIN 1.0f
#define KAPPA_MAX 100000.0f

typedef __attribute__((ext_vector_type(2))) float v2f;
typedef __attribute__((ext_vector_type(8))) float v8f;

// ---------------------------------------------------------------------------
// log(i0e(x)) for x >= 0, Abramowitz & Stegun 9.8.1 / 9.8.2 (fp32)
// ---------------------------------------------------------------------------
__device__ __forceinline__ float log_i0e(float x) {
  if (x < 3.75f) {
    float t = x * (1.0f / 3.75f);
    t = t * t;
    float p = 1.0f + t * (3.5156229f + t * (3.0899424f + t * (1.2067492f +
              t * (0.2659732f + t * (0.0360768f + t * 0.0045813f)))));
    return __logf(p) - x;              // i0e = i0 * exp(-x)
  } else {
    float u = 3.75f / x;
    float p = 0.39894228f + u * (0.01328592f + u * (0.00225319f + u * (-0.00157565f +
              u * (0.00916281f + u * (-0.02057706f + u * (0.02635537f +
              u * (-0.01647633f + u * 0.00392377f)))))));
    return __logf(p) - 0.5f * __logf(x);   // i0e = p / sqrt(x)
  }
}

// ---------------------------------------------------------------------------
// Scalar Miller backward recurrence: log(ive(NU, x)) — mirrors reference scan
// ---------------------------------------------------------------------------
__device__ __forceinline__ float log_ive_nu(float x) {
  x = fmaxf(x, 1e-20f);
  float rx = 1.0f / x;
  float b = 0.0f, scale = 0.0f, e0 = 1.0f, s0 = 0.0f;
  for (int i = 2 * NU; i >= 1; --i) {
    float s = fmaf(2.0f * (float)i, rx, b);   // 2i/x + b
    b = __builtin_amdgcn_rcpf(s);             // normalized I_i = 1/s
    scale += __logf(s);
    if (i == NU) { e0 = b; s0 = scale; }
  }
  return log_i0e(x) + __logf(e0) + s0 - scale;
}

// ---------------------------------------------------------------------------
// Kernel 1: Nmin / Nmax of class_freq (deterministic LDS tree)
// ---------------------------------------------------------------------------
__global__ __launch_bounds__(1024) void k_minmax(const int* __restrict__ freq,
                                                 float* __restrict__ nmm) {
  __shared__ int smin[1024], smax[1024];
  int t = threadIdx.x;
  int mn = 0x7fffffff, mx = (int)0x80000000;
  for (int i = t; i < C_CLS; i += 1024) {
    int v = freq[i];
    mn = min(mn, v);
    mx = max(mx, v);
  }
  smin[t] = mn; smax[t] = mx;
  __syncthreads();
  for (int off = 512; off > 0; off >>= 1) {
    if (t < off) { smin[t] = min(smin[t], smin[t + off]); smax[t] = max(smax[t], smax[t + off]); }
    __syncthreads();
  }
  if (t == 0) { nmm[0] = (float)smin[0]; nmm[1] = (float)smax[0]; }
}

// ---------------------------------------------------------------------------
// Kernel 2: ||f_n||^2 per batch row
// ---------------------------------------------------------------------------
__global__ __launch_bounds__(128) void k_fnorm(const float* __restrict__ feats,
                                               float* __restrict__ fnorm2) {
  int n = blockIdx.x, p = threadIdx.x;
  __shared__ float red[128];
  float f = feats[n * P_DIM + p];
  red[p] = f * f;
  __syncthreads();
  for (int off = 64; off > 0; off >>= 1) {
    if (p < off) red[p] += red[p + off];
    __syncthreads();
  }
  if (p == 0) fnorm2[n] = red[0];
}

// ---------------------------------------------------------------------------
// Kernel 3: per-class statistics -> kappa, logc, normalized Ave, ||Ave_n||^2
// Deterministic gather (no atomics): one block per class scans the batch.
// ---------------------------------------------------------------------------
__global__ __launch_bounds__(128) void k_class(const float* __restrict__ feats,
                                               const int* __restrict__ labels,
                                               const float* __restrict__ Ave0,
                                               const int* __restrict__ freq,
                                               const float* __restrict__ nmm,
                                               float* __restrict__ kappa,
                                               float* __restrict__ logc,
                                               float* __restrict__ anrm2,
                                               float* __restrict__ aveN) {
  int c = blockIdx.x, p = threadIdx.x;
  __shared__ int   slab[N_BATCH];
  __shared__ float red[128];
  for (int i = p; i < N_BATCH; i += 128) slab[i] = labels[i];
  __syncthreads();

  float sum = 0.0f, cnt = 0.0f;
  for (int n = 0; n < N_BATCH; ++n) {
    if (slab[n] == c) { sum += feats[n * P_DIM + p]; cnt += 1.0f; }
  }
  float ave = sum / fmaxf(cnt, 1.0f);
  float A   = (cnt > 0.0f) ? ave : Ave0[c * P_DIM + p];

  red[p] = A * A;
  __syncthreads();
  for (int off = 64; off > 0; off >>= 1) {
    if (p < off) red[p] += red[p + off];
    __syncthreads();
  }
  float R2 = red[0];
  float R  = sqrtf(R2);

  float rden = fmaxf(R, 1e-12f);
  aveN[c * P_DIM + p] = A / rden;

  if (p == 0) {
    float k0 = 128.0f * R / (1.0f - R2);
    if (k0 > KAPPA_MAX || k0 < 0.0f) k0 = KAPPA_MAX;
    float Ny = (float)freq[c];
    float Nmin = nmm[0], Nmax = nmm[1];
    float adj = (Nmax == Nmin) ? 1.0f
                               : sqrtf(1.0f - (Ny - Nmin) / (Nmax - Nmin)); // BETA=0.5
    float kap = KAPPA_MIN + (k0 - KAPPA_MIN) * adj;
    kap = fminf(fmaxf(kap, KAPPA_MIN), KAPPA_MAX);
    float lnu = log_ive_nu(kap);
    kappa[c] = kap;
    logc[c]  = lnu + kap - 63.0f * __logf(kap + 1e-20f);
    float rn = R / rden;          // 1 unless ||Ave|| < 1e-12
    anrm2[c] = rn * rn;
  }
}

// ---------------------------------------------------------------------------
// Kernel 4: main. One wave per 16x16 output tile.
//   D = feats(512x128) x aveN^T(128x1000) via v_wmma_f32_16x16x4_f32
//   then kn = sqrt(k^2*||An||^2 + 2k*D + ||f||^2), Miller recurrence epilogue
// fp32 WMMA fragment layout (ISA 7.12.2):
//   A 16x4 : lane L, vgpr v -> M = L&15,  K = v + 2*(L>>4)
//   B 4x16 : lane L, vgpr v -> N = L&15,  K = v + 2*(L>>4)
//   C 16x16: lane L, vgpr v -> N = L&15,  M = v + 8*(L>>4)
// Both A and B fragments are a float2 load at row*128 + k + 2*(L>>4).
// ---------------------------------------------------------------------------
__global__ __launch_bounds__(32) void k_main(const float* __restrict__ feats,
                                             const float* __restrict__ aveN,
                                             const float* __restrict__ kappa,
                                             const float* __restrict__ logc,
                                             const float* __restrict__ anrm2,
                                             const float* __restrict__ fnorm2,
                                             float* __restrict__ out) {
  const int lane = threadIdx.x;
  const int row0 = blockIdx.x * 16;
  const int col0 = blockIdx.y * 16;
  const int m     = lane & 15;
  const int hi    = lane >> 4;
  const int khalf = hi * 2;

  const int cls  = col0 + m;                      // B/N index == lane&15
  const int clsc = (cls < C_CLS) ? cls : (C_CLS - 1);

  const float* arow = feats + (row0 + m) * P_DIM + khalf;
  const float* brow = aveN + clsc * P_DIM + khalf;
  __builtin_prefetch(arow, 0, 0);
  __builtin_prefetch(brow, 0, 0);

  v8f acc = {};
#pragma unroll
  for (int k = 0; k < P_DIM; k += 4) {
    v2f a = *(const v2f*)(arow + k);
    v2f b = *(const v2f*)(brow + k);
    // D = A * B + C  (fp32, 16x16x4)
    acc = __builtin_amdgcn_wmma_f32_16x16x4_f32(false, a, false, b,
                                                (short)0, acc, false, false);
  }

  const float kc  = kappa[clsc];
  const float lc  = logc[clsc];
  const float an2 = anrm2[clsc];

  // kn for the 8 outputs this lane owns
  float kn[8];
#pragma unroll
  for (int v = 0; v < 8; ++v) {
    int row = row0 + v + 8 * hi;
    float d   = acc[v];
    float kn2 = fmaf(kc * kc, an2, fmaf(2.0f * kc, d, fnorm2[row]));
    kn[v] = sqrtf(fmaxf(kn2, 0.0f));
  }

  // 8-wide Miller backward recurrence (ILP hides v_log/v_rcp latency)
  float b8[8], scale[8], e0[8], s0[8], rx[8];
#pragma unroll
  for (int v = 0; v < 8; ++v) {
    b8[v] = 0.0f; scale[v] = 0.0f; e0[v] = 1.0f; s0[v] = 0.0f;
    rx[v] = 1.0f / fmaxf(kn[v], 1e-20f);
  }
  for (int i = 2 * NU; i >= 1; --i) {
    float twoi = 2.0f * (float)i;
#pragma unroll
    for (int v = 0; v < 8; ++v) {
      float s = fmaf(twoi, rx[v], b8[v]);
      b8[v] = __builtin_amdgcn_rcpf(s);
      scale[v] += __logf(s);
    }
    if (i == NU) {
#pragma unroll
      for (int v = 0; v < 8; ++v) { e0[v] = b8[v]; s0[v] = scale[v]; }
    }
  }

  if (cls < C_CLS) {
#pragma unroll
    for (int v = 0; v < 8; ++v) {
      int row = row0 + v + 8 * hi;
      float x   = kn[v];
      float lnu = log_i0e(x) + __logf(e0[v]) + s0[v] - scale[v];
      out[row * C_CLS + cls] = lnu + x - 63.0f * __logf(x + 1e-20f) - lc;
    }
  }
}

// ---------------------------------------------------------------------------
extern "C" void kernel_launch(void* const* d_in, const int* in_sizes, int n_in,
                              void* d_out, int out_size, void* d_ws, size_t ws_size,
                              hipStream_t stream) {
  const float* feats  = (const float*)d_in[0];   // [512,128]
  const int*   labels = (const int*)d_in[1];     // [512]
  const float* Ave0   = (const float*)d_in[2];   // [1000,128]
  const int*   freq   = (const int*)d_in[3];     // [1000]
  float* out = (float*)d_out;                    // [512,1000]

  float* ws     = (float*)d_ws;
  float* kappa  = ws;            // 1000
  float* logc   = ws + 1000;     // 1000
  float* anrm2  = ws + 2000;     // 1000
  float* nmm    = ws + 3000;     // 2 (pad to 8)
  float* fnorm2 = ws + 3008;     // 512
  float* aveN   = ws + 3520;     // 1000*128 (8B-aligned for float2 loads)

  k_minmax<<<dim3(1), dim3(1024), 0, stream>>>(freq, nmm);
  k_fnorm<<<dim3(N_BATCH), dim3(128), 0, stream>>>(feats, fnorm2);
  k_class<<<dim3(C_CLS), dim3(128), 0, stream>>>(feats, labels, Ave0, freq, nmm,
                                                 kappa, logc, anrm2, aveN);
  k_main<<<dim3(N_BATCH / 16, (C_CLS + 15) / 16), dim3(32), 0, stream>>>(
      feats, aveN, kappa, logc, anrm2, fnorm2, out);
}